// IdentityGNN_71579924955534
// MI455X (gfx1250) — compile-verified
//
#include <hip/hip_runtime.h>
#include <hip/hip_bf16.h>

#define N_NODES 100000
#define N_EDGES 1250000
#define IN_DIM  12
#define HID     64
#define N_LAYERS 3

typedef __attribute__((ext_vector_type(2))) float v2f;
typedef __attribute__((ext_vector_type(8))) float v8f;

// ---------------------------------------------------------------- input proj
// h[n][c] = sum_k x[n][k] * in_W[k][c] + in_b[c]   (K=12, memory-bound)
__global__ void input_proj_kernel(const float* __restrict__ x,
                                  const float* __restrict__ W,
                                  const float* __restrict__ b,
                                  float* __restrict__ h) {
    int idx = blockIdx.x * blockDim.x + threadIdx.x;   // N*64 threads exactly
    int n = idx >> 6, c = idx & 63;
    const float* xp = x + (size_t)n * IN_DIM;
    float acc = b[c];
#pragma unroll
    for (int k = 0; k < IN_DIM; ++k) acc += xp[k] * W[k * HID + c];
    h[idx] = acc;
}

// ------------------------------------------------------ per-layer init (agg=0)
__global__ void layer_init_kernel(float* __restrict__ agg,
                                  unsigned* __restrict__ redmax,
                                  float* __restrict__ redsum) {
    int idx = blockIdx.x * blockDim.x + threadIdx.x;   // N*64 threads exactly
    agg[idx] = 0.0f;
    if (idx == 0) { *redmax = 0u; *redsum = 0.0f; }
}

// ----------------------------------------------------- hl = h @ W + b (WMMA)
// fp32 GEMM via V_WMMA_F32_16X16X4_F32. One wave -> one 16x16 output tile,
// K-loop of 16 chained WMMAs. Block = 4 waves = 16 rows x 64 cols.
// N_NODES/16 = 6250 blocks exactly -> EXEC all-ones everywhere.
__global__ void gemm64_wmma_kernel(const float* __restrict__ A,   // [N,64]
                                   const float* __restrict__ W,   // [64,64]
                                   const float* __restrict__ bias,// [64]
                                   float* __restrict__ out) {     // [N,64]
    const int lane = threadIdx.x & 31;
    const int wave = threadIdx.x >> 5;          // 0..3 -> column tile
    const int m0   = blockIdx.x << 4;           // row-tile base
    const int n0   = wave << 4;
    const int mrow = lane & 15;                 // A: M index
    const int ncol = lane & 15;                 // B/C: N index
    const int ksub = (lane >> 4) << 1;          // 0 (lanes 0-15) or 2 (16-31)

    v8f acc = {};
#pragma unroll
    for (int k = 0; k < HID; k += 4) {
        v2f a, b;
        const float* ap = A + (size_t)(m0 + mrow) * HID + (k + ksub);
        a.x = ap[0];
        a.y = ap[1];
        const float* bp = W + (size_t)(k + ksub) * HID + n0 + ncol;
        b.x = bp[0];
        b.y = bp[HID];
        acc = __builtin_amdgcn_wmma_f32_16x16x4_f32(
            /*neg_a=*/false, a, /*neg_b=*/false, b,
            /*c_mod=*/(short)0, acc, /*reuse_a=*/false, /*reuse_b=*/false);
    }
    const float bv  = bias[n0 + ncol];
    const int   mhi = (lane >> 4) << 3;         // 0 or 8
#pragma unroll
    for (int r = 0; r < 8; ++r)
        out[(size_t)(m0 + r + mhi) * HID + n0 + ncol] = acc[r] + bv;
}

// ----------------------------------------- per-node attention half-dot-products
// a[n] = hl[n,:]·attW[0:64], b[n] = hl[n,:]·attW[64:128]
__global__ void node_att_kernel(const float* __restrict__ hl,
                                const float* __restrict__ attW,
                                float* __restrict__ an, float* __restrict__ bn) {
    int n = blockIdx.x * blockDim.x + threadIdx.x;
    if (n >= N_NODES) return;
    const float* hp = hl + (size_t)n * HID;
    float a = 0.f, b = 0.f;
#pragma unroll
    for (int k = 0; k < HID; ++k) {
        float v = hp[k];
        a += v * attW[k];
        b += v * attW[HID + k];
    }
    an[n] = a; bn[n] = b;
}

// ----------------------------------------------- softmax pass 1: score + max
__global__ void edge_score_max_kernel(const int* __restrict__ ei,
                                      const float* __restrict__ an,
                                      const float* __restrict__ bn,
                                      const float* __restrict__ attb,
                                      float* __restrict__ escore,
                                      unsigned* __restrict__ redmax) {
    int e = blockIdx.x * blockDim.x + threadIdx.x;
    float s = -3.4e38f;
    if (e < N_EDGES) {
        int r = ei[e], c = ei[N_EDGES + e];
        float x = an[r] + bn[c] + attb[0];
        s = (x > 0.f) ? x : 0.01f * x;          // leaky_relu
        escore[e] = s;
    }
#pragma unroll
    for (int m = 16; m; m >>= 1) s = fmaxf(s, __shfl_xor(s, m, 32));
    if ((threadIdx.x & 31) == 0) {
        unsigned u   = __float_as_uint(s);
        unsigned key = (u >> 31) ? ~u : (u | 0x80000000u); // order-preserving
        atomicMax(redmax, key);
    }
}

// ----------------------------------------------- softmax pass 2: exp + sum
__global__ void edge_expsum_kernel(float* __restrict__ escore,
                                   const unsigned* __restrict__ redmax,
                                   float* __restrict__ redsum) {
    int e = blockIdx.x * blockDim.x + threadIdx.x;
    unsigned key = *redmax;
    unsigned u   = (key & 0x80000000u) ? (key & 0x7FFFFFFFu) : ~key;
    float mx     = __uint_as_float(u);
    float v = 0.f;
    if (e < N_EDGES) {
        v = __expf(escore[e] - mx);
        escore[e] = v;                          // now holds exp(score - max)
    }
#pragma unroll
    for (int m = 16; m; m >>= 1) v += __shfl_xor(v, m, 32);
    if ((threadIdx.x & 31) == 0) atomicAdd(redsum, v);
}

// ------------------------------- message scatter: agg[row] += coef * hl[col]
// 16 lanes per edge, float4 per lane, 4 atomic f32 adds (L2-resident).
__global__ void edge_scatter_kernel(const int* __restrict__ ei,
                                    const float* __restrict__ ew,
                                    const float* __restrict__ escore,
                                    const float* __restrict__ redsum,
                                    const float* __restrict__ hl,
                                    float* __restrict__ agg) {
    int t = blockIdx.x * blockDim.x + threadIdx.x;   // N_EDGES*16 threads exact
    int e = t >> 4, sub = t & 15;
    float inv  = 1.0f / redsum[0];
    float coef = escore[e] * ew[e] * inv;
    int r = ei[e], c = ei[N_EDGES + e];
    const float4 hv = *(const float4*)(hl + (size_t)c * HID + sub * 4);
    float* ap = agg + (size_t)r * HID + sub * 4;
    atomicAdd(ap + 0, coef * hv.x);
    atomicAdd(ap + 1, coef * hv.y);
    atomicAdd(ap + 2, coef * hv.z);
    atomicAdd(ap + 3, coef * hv.w);
}

// --------------------------- h += layernorm(relu(hl + agg)) * g + b  (wave/node)
__global__ void node_update_kernel(const float* __restrict__ hl,
                                   const float* __restrict__ agg,
                                   const float* __restrict__ g,
                                   const float* __restrict__ b,
                                   float* __restrict__ h) {
    int wid  = (blockIdx.x * blockDim.x + threadIdx.x) >> 5;  // node
    int lane = threadIdx.x & 31;
    size_t base = (size_t)wid * HID;
    float v0 = fmaxf(hl[base + lane]      + agg[base + lane],      0.f);
    float v1 = fmaxf(hl[base + lane + 32] + agg[base + lane + 32], 0.f);
    float s = v0 + v1;
#pragma unroll
    for (int m = 16; m; m >>= 1) s += __shfl_xor(s, m, 32);
    float mu = s * (1.0f / HID);
    float d0 = v0 - mu, d1 = v1 - mu;
    float q = d0 * d0 + d1 * d1;
#pragma unroll
    for (int m = 16; m; m >>= 1) q += __shfl_xor(q, m, 32);
    float rs = rsqrtf(q * (1.0f / HID) + 1e-5f);
    h[base + lane]      += d0 * rs * g[lane]      + b[lane];
    h[base + lane + 32] += d1 * rs * g[lane + 32] + b[lane + 32];
}

// ---------------- output head: sigmoid(relu(h@W1+b1)@W2+b2)  (wave/node)
__global__ void out_head_kernel(const float* __restrict__ h,
                                const float* __restrict__ W1,
                                const float* __restrict__ b1,
                                const float* __restrict__ W2,
                                const float* __restrict__ b2,
                                float* __restrict__ out) {
    int wid  = (blockIdx.x * blockDim.x + threadIdx.x) >> 5;  // node
    int lane = threadIdx.x & 31;                              // hidden unit (32)
    const float* hp = h + (size_t)wid * HID;
    float z = b1[lane];
#pragma unroll
    for (int k = 0; k < HID; ++k) z += hp[k] * W1[k * 32 + lane];
    z = fmaxf(z, 0.f);
    float p = z * W2[lane];
#pragma unroll
    for (int m = 16; m; m >>= 1) p += __shfl_xor(p, m, 32);
    if (lane == 0) out[wid] = 1.0f / (1.0f + __expf(-(p + b2[0])));
}

extern "C" void kernel_launch(void* const* d_in, const int* in_sizes, int n_in,
                              void* d_out, int out_size, void* d_ws, size_t ws_size,
                              hipStream_t stream) {
    const float* x      = (const float*)d_in[0];
    const int*   ei     = (const int*)  d_in[1];   // edge_index [2,E]
    const float* ew     = (const float*)d_in[2];
    const float* in_W   = (const float*)d_in[3];
    const float* in_b   = (const float*)d_in[4];
    const float* lin_W  = (const float*)d_in[5];   // [3,64,64]
    const float* lin_b  = (const float*)d_in[6];   // [3,64]
    const float* att_W  = (const float*)d_in[7];   // [3,128]
    const float* att_b  = (const float*)d_in[8];   // [3]
    const float* ln_g   = (const float*)d_in[9];   // [3,64]
    const float* ln_b   = (const float*)d_in[10];  // [3,64]
    const float* out1_W = (const float*)d_in[11];
    const float* out1_b = (const float*)d_in[12];
    const float* out2_W = (const float*)d_in[13];
    const float* out2_b = (const float*)d_in[14];

    const size_t NH = (size_t)N_NODES * HID;       // 6.4M floats
    float* ws     = (float*)d_ws;
    float* h      = ws;                            // [N,64]
    float* hl     = ws + NH;                       // [N,64]
    float* agg    = ws + 2 * NH;                   // [N,64]
    float* an     = ws + 3 * NH;                   // [N]
    float* bn     = an + N_NODES;                  // [N]
    float* escore = bn + N_NODES;                  // [E]
    float*    red    = escore + N_EDGES;           // [2]
    unsigned* redmax = (unsigned*)red;
    float*    redsum = red + 1;

    const int elemBlocks = (int)(NH / 256);                       // 25000 exact
    const int gemmBlocks = N_NODES / 16;                          // 6250 exact
    const int nodeBlocks = (N_NODES + 255) / 256;                 // 391
    const int waveBlocks = N_NODES / 8;                           // 12500 exact (8 waves/block)
    const int edgeBlocks = (N_EDGES + 255) / 256;                 // 4883
    const int scatBlocks = (N_EDGES * 16) / 256;                  // 78125 exact

    input_proj_kernel<<<elemBlocks, 256, 0, stream>>>(x, in_W, in_b, h);

    for (int L = 0; L < N_LAYERS; ++L) {
        layer_init_kernel<<<elemBlocks, 256, 0, stream>>>(agg, redmax, redsum);
        gemm64_wmma_kernel<<<gemmBlocks, 128, 0, stream>>>(
            h, lin_W + (size_t)L * HID * HID, lin_b + (size_t)L * HID, hl);
        node_att_kernel<<<nodeBlocks, 256, 0, stream>>>(
            hl, att_W + (size_t)L * 2 * HID, an, bn);
        edge_score_max_kernel<<<edgeBlocks, 256, 0, stream>>>(
            ei, an, bn, att_b + L, escore, redmax);
        edge_expsum_kernel<<<edgeBlocks, 256, 0, stream>>>(escore, redmax, redsum);
        edge_scatter_kernel<<<scatBlocks, 256, 0, stream>>>(
            ei, ew, escore, redsum, hl, agg);
        node_update_kernel<<<waveBlocks, 256, 0, stream>>>(
            hl, agg, ln_g + (size_t)L * HID, ln_b + (size_t)L * HID, h);
    }

    out_head_kernel<<<waveBlocks, 256, 0, stream>>>(
        h, out1_W, out1_b, out2_W, out2_b, (float*)d_out);
}